// HTLoss_56298431316040
// MI455X (gfx1250) — compile-verified
//
#include <hip/hip_runtime.h>

typedef float v2f __attribute__((ext_vector_type(2)));
typedef float v8f __attribute__((ext_vector_type(8)));

#define H_HT 26
#define W_HT 122
#define NUM_ANGLE 60
#define NUM_RHO 125
#define NB 32
#define NC 4
#define M_ (NB*NC)              // 128
#define K_ (H_HT*W_HT)          // 3172
#define KP 3200                 // K padded to multiple of K_BLK (pad is ZERO in A)
#define N_ (NUM_ANGLE*NUM_RHO)  // 7500
#define K_BLK 64
#define N_BLK 32
#define NT_BLKS ((N_ + N_BLK - 1) / N_BLK)   // 235
#define N_PAD (NT_BLKS * N_BLK)              // 7520
#define NPANELS (KP / K_BLK)                 // 50
#define PANEL_ELTS (K_BLK * N_BLK)           // 2048 floats
#define PRED_H 288
#define PRED_W 800

// ---------------------------------------------------------------------------
// 1) Nearest downsample pred -> A [128, KP] f32, zero-filled K pad.
//    hi = floor(h*288/26), wi = floor(w*800/122) (exact in integer arithmetic)
// ---------------------------------------------------------------------------
__global__ void __launch_bounds__(256)
ht_downsample(const float* __restrict__ pred, float* __restrict__ A) {
    int o = blockIdx.x * blockDim.x + threadIdx.x;
    if (o >= M_ * KP) return;
    int bc = o / KP;
    int k  = o - bc * KP;
    float v = 0.0f;
    if (k < K_) {
        int h  = k / W_HT;
        int w  = k - h * W_HT;
        int hi = (h * PRED_H) / H_HT;
        int wi = (w * PRED_W) / W_HT;
        v = pred[(size_t)bc * (PRED_H * PRED_W) + hi * PRED_W + wi];
    }
    A[o] = v;
}

// ---------------------------------------------------------------------------
// Swizzled LDS index for B panel element (row r in [0,64), col c in [0,32)):
//   - K-row pairs (2g, 2g+1) adjacent -> one ds_load_b64 per WMMA fragment
//   - 32-slot parity skew per pair-group -> the two half-waves (pair groups
//     g and g+1, opposite parity) use disjoint bank halves: conflict-free
// ---------------------------------------------------------------------------
__device__ __forceinline__ int panel_idx(int r, int c) {
    int g = r >> 1;
    return (g << 6) + ((((c << 1) | (r & 1)) + ((g & 1) << 5)) & 63);
}

// ---------------------------------------------------------------------------
// Async copy of one B panel (K_BLK x N_BLK f32) into swizzled LDS.
// 128 threads x 16 b32 transfers (lanes read 128 contiguous global bytes).
// Addresses clamped in-bounds:
//  - clamped K rows only feed zero A-pad columns (A pad == 0)
//  - clamped N cols only affect padded output columns >= 7500 (never read)
// ---------------------------------------------------------------------------
__device__ __forceinline__ void
load_panel_async(const float* __restrict__ Bm, float* dst, int k0, int n_base, int tid) {
    #pragma unroll
    for (int i = 0; i < PANEL_ELTS / 128; ++i) {   // 16 iters
        int e = tid + 128 * i;
        int r = e >> 5;             // 0..63
        int c = e & 31;
        int gcol = n_base + c;
        gcol = gcol > (N_ - 1) ? (N_ - 1) : gcol;
        int grow = k0 + r;
        grow = grow > (K_ - 1) ? (K_ - 1) : grow;
        const float* gaddr = Bm + (size_t)grow * N_ + gcol;
        unsigned ldsoff = (unsigned)(uintptr_t)(dst + panel_idx(r, c));
        asm volatile("global_load_async_to_lds_b32 %0, %1, off"
                     :: "v"(ldsoff), "v"(gaddr) : "memory");
    }
}

// ---------------------------------------------------------------------------
// 2) C[128, N_PAD] = A[128, KP] x vote[K_, N_] via v_wmma_f32_16x16x4_f32.
//    Block = 4 wave32, 32-wide N strip, all of M. B panels double-buffered
//    in LDS via async loads; each wave owns 2 M tiles x 2 N tiles.
//    Operand layouts per CDNA5 ISA 7.12.2.
// ---------------------------------------------------------------------------
__global__ void __launch_bounds__(128)
ht_vote_gemm(const float* __restrict__ A,
             const float* __restrict__ Bm,
             float* __restrict__ Cp) {
    __shared__ float lds[2][PANEL_ELTS];   // 2 x 8 KB

    const int tid  = threadIdx.x;
    const int wave = tid >> 5;     // 0..3 -> M tiles 2w, 2w+1
    const int lane = tid & 31;
    const int half = lane >> 4;    // K sub-pair select
    const int lm   = lane & 15;
    const int n_base = blockIdx.x * N_BLK;
    const int mt0  = wave * 2;

    v8f acc[2][2] = {};

    load_panel_async(Bm, &lds[0][0], 0, n_base, tid);
    asm volatile("s_wait_asynccnt 0" ::: "memory");
    __syncthreads();

    int cur = 0;
    for (int panel = 0; panel < NPANELS; ++panel) {
        if (panel + 1 < NPANELS)
            load_panel_async(Bm, &lds[cur ^ 1][0], (panel + 1) * K_BLK, n_base, tid);

        const float* lp = &lds[cur][0];
        const int k0 = panel * K_BLK;
        #pragma unroll 4
        for (int kk = 0; kk < K_BLK; kk += 4) {
            const int g0 = (kk >> 1) + half;            // pair-group for this half
            const int p  = (g0 & 1) << 5;               // parity skew
            v2f b0 = *(const v2f*)(lp + (g0 << 6) + (((lm << 1) + p) & 63));
            v2f b1 = *(const v2f*)(lp + (g0 << 6) + (((lm << 1) + 32 + p) & 63));
            #pragma unroll
            for (int u = 0; u < 2; ++u) {
                const int row = (mt0 + u) * 16 + lm;
                v2f a = *(const v2f*)(A + (size_t)row * KP + k0 + kk + 2 * half);
                acc[u][0] = __builtin_amdgcn_wmma_f32_16x16x4_f32(
                    false, a, false, b0, (short)0, acc[u][0], false, false);
                acc[u][1] = __builtin_amdgcn_wmma_f32_16x16x4_f32(
                    false, a, false, b1, (short)0, acc[u][1], false, false);
            }
        }
        asm volatile("s_wait_asynccnt 0" ::: "memory");
        __syncthreads();
        cur ^= 1;
    }

    #pragma unroll
    for (int u = 0; u < 2; ++u)
        #pragma unroll
        for (int nt = 0; nt < 2; ++nt)
            #pragma unroll
            for (int v = 0; v < 8; ++v) {
                const int rowg = (mt0 + u) * 16 + v + 8 * half;
                Cp[(size_t)rowg * N_PAD + n_base + nt * 16 + lm] = acc[u][nt][v];
            }
}

// ---------------------------------------------------------------------------
// 3) Per-row argmax over 7500 bins (first-occurrence tie-break, like jnp)
// ---------------------------------------------------------------------------
__global__ void __launch_bounds__(256)
ht_argmax(const float* __restrict__ Cp, int* __restrict__ idx) {
    __shared__ float sv[256];
    __shared__ int   si[256];
    const int row = blockIdx.x;
    const float* r = Cp + (size_t)row * N_PAD;
    float best = -__builtin_inff();
    int   bi   = 0;
    for (int i = threadIdx.x; i < N_; i += 256) {
        float v = r[i];
        if (v > best) { best = v; bi = i; }
    }
    sv[threadIdx.x] = best;
    si[threadIdx.x] = bi;
    __syncthreads();
    for (int s = 128; s > 0; s >>= 1) {
        if (threadIdx.x < s) {
            float ov = sv[threadIdx.x + s];
            int   oi = si[threadIdx.x + s];
            if (ov > sv[threadIdx.x] ||
                (ov == sv[threadIdx.x] && oi < si[threadIdx.x])) {
                sv[threadIdx.x] = ov;
                si[threadIdx.x] = oi;
            }
        }
        __syncthreads();
    }
    if (threadIdx.x == 0) idx[row] = si[0];
}

// ---------------------------------------------------------------------------
// 4) Final loss: L1-normalize ht along angle at argmax rho, gather,
//    -log(.+1e-12) * exist, mean over (c,j) -> out[b]
// ---------------------------------------------------------------------------
__global__ void __launch_bounds__(32)
ht_loss(const float* __restrict__ ht,
        const float* __restrict__ pred_exist,
        const int* __restrict__ idx,
        float* __restrict__ out) {
    __shared__ float s[32];
    const int b = blockIdx.x;
    const int t = threadIdx.x;
    float term = 0.0f;
    if (t < 16) {
        const int c = t >> 2;
        const int j = t & 3;
        const int iflat = idx[b * NC + j];
        const int a  = iflat / NUM_RHO;
        const int rr = iflat - a * NUM_RHO;
        const float* hb = ht + (size_t)(b * NC + c) * (NUM_ANGLE * NUM_RHO);
        float denom = 0.0f;
        for (int a2 = 0; a2 < NUM_ANGLE; ++a2)
            denom += fabsf(hb[a2 * NUM_RHO + rr]);
        denom = fmaxf(denom, 1e-12f);
        const float val = hb[a * NUM_RHO + rr] / denom;
        const float ex  = (pred_exist[b * NC + j] > 0.9f) ? 1.0f : 0.0f;
        term = -logf(val + 1e-12f) * ex;
    }
    s[t] = term;
    __syncthreads();
    if (t == 0) {
        float sum = 0.0f;
        for (int u = 0; u < 16; ++u) sum += s[u];
        out[b] = sum * (1.0f / 16.0f);
    }
}

// ---------------------------------------------------------------------------
extern "C" void kernel_launch(void* const* d_in, const int* in_sizes, int n_in,
                              void* d_out, int out_size, void* d_ws, size_t ws_size,
                              hipStream_t stream) {
    const float* ht         = (const float*)d_in[0];  // [32,4,60,125]
    const float* pred       = (const float*)d_in[1];  // [32,4,288,800]
    const float* pred_exist = (const float*)d_in[2];  // [32,4]
    const float* vote       = (const float*)d_in[3];  // [3172, 7500]
    float* out = (float*)d_out;                       // [32]

    char* ws = (char*)d_ws;
    float* A_s  = (float*)ws;                              // 128*3200*4 = 1,638,400 B
    float* C_ht = (float*)(ws + 1638400);                  // 128*7520*4 = 3,850,240 B
    int*   idx  = (int*)  (ws + 1638400 + 3850240);        // 512 B

    ht_downsample<<<(M_ * KP + 255) / 256, 256, 0, stream>>>(pred, A_s);
    ht_vote_gemm<<<NT_BLKS, 128, 0, stream>>>(A_s, vote, C_ht);
    ht_argmax<<<M_, 256, 0, stream>>>(C_ht, idx);
    ht_loss<<<NB, 32, 0, stream>>>(ht, pred_exist, idx, out);
}